// PercentileAggregator_18184891531885
// MI455X (gfx1250) — compile-verified
//
#include <hip/hip_runtime.h>
#include <stdint.h>

#define NPCT    10
#define NRANK   20          // lo+hi rank per percentile, all distinct here
#define NN      16384       // N (column length)
#define DD      256         // D
#define BB      16          // B
#define CAP     256         // candidate capacity per 16-bit-prefix slot
#define THREADS 256
#define ITERS   (NN / THREADS)   // 64 per-thread elements

// ---- CDNA5 async global->LDS path (probe via __has_builtin; safe fallback) ----
#if defined(__AMDGCN__) && \
    __has_builtin(__builtin_amdgcn_global_load_async_to_lds_b32) && \
    __has_builtin(__builtin_amdgcn_s_wait_asynccnt)
#define USE_ASYNC_LDS 1
typedef __attribute__((address_space(3))) int lds_int;
typedef __attribute__((address_space(1))) int gbl_int;
#else
#define USE_ASYNC_LDS 0
#endif

// Order-preserving monotone key for IEEE-754 float bits.
__device__ __forceinline__ unsigned keyOf(unsigned u) {
  return u ^ (((unsigned)((int)u >> 31)) | 0x80000000u);
}
__device__ __forceinline__ float fromKey(unsigned k) {
  unsigned m = ((unsigned)((int)(~k) >> 31)) | 0x80000000u;
  return __uint_as_float(k ^ m);
}

__global__ __launch_bounds__(THREADS)
void PercentileAggregator_kernel(const float* __restrict__ x, float* __restrict__ out)
{
  __shared__ unsigned data[NN];          // raw float bits of the column (64 KB)
  __shared__ unsigned H1[256];           // level-1 histogram (key[31:24])
  __shared__ int      map1[256];         // byte -> slot (or -1)
  __shared__ unsigned H2[NRANK * 256];   // level-2 histograms per slot (20 KB)
  __shared__ unsigned cand[NRANK * CAP]; // candidate keys per 16-bit prefix slot
  __shared__ unsigned cnt2[NRANK];
  __shared__ int      slotOfRank[NRANK], rank1[NRANK];
  __shared__ int      bucket1OfSlot[NRANK];
  __shared__ unsigned prefix2[NRANK];
  __shared__ int      slot2OfRank[NRANK], rank2[NRANK];
  __shared__ int      nslot2;
  __shared__ unsigned keyOut[NRANK];

  const int tid = threadIdx.x;
  const int col = blockIdx.x;        // 0..4095
  const int b   = col >> 8;          // D = 256
  const int d   = col & 255;
  const size_t base = ((size_t)b * NN) * DD + (size_t)d;

  // Histogram one quarter of this thread's own async-loaded elements.
  auto hist16 = [&](int i0) {
    #pragma clang loop unroll_count(4)
    for (int i = i0; i < i0 + (ITERS / 4); ++i) {
      unsigned key = keyOf(data[i * THREADS + tid]);
      atomicAdd(&H1[key >> 24], 1u);
    }
  };

  // Zero counters first (barrier below makes them visible before any atomics).
  H1[tid] = 0u;
  if (tid < NRANK) { cnt2[tid] = 0u; keyOut[tid] = 0u; }

  // ---------------- Phase A: stage column into LDS (raw bits) ----------------
#if USE_ASYNC_LDS
  for (int i = 0; i < ITERS; ++i) {
    int n = i * THREADS + tid;
    const float* gp = x + base + (size_t)n * DD;
    // AS casts via 32/64-bit integers: LDS offset lives in low 32 bits.
    gbl_int* g  = (gbl_int*)(uintptr_t)gp;
    lds_int* lp = (lds_int*)(unsigned)(uintptr_t)(&data[n]);
    __builtin_amdgcn_global_load_async_to_lds_b32(g, lp, 0, 0);
  }
  __syncthreads();                       // H1 zeroed everywhere

  // ------- Phase B overlapped with load latency (in-order completion) -------
  // Each wave consumes only elements it issued itself, so per-wave ASYNCcnt
  // thresholds are a valid readiness guarantee for its own LDS slots.
  __builtin_amdgcn_s_wait_asynccnt(48);  hist16(0);
  __builtin_amdgcn_s_wait_asynccnt(32);  hist16(16);
  __builtin_amdgcn_s_wait_asynccnt(16);  hist16(32);
  __builtin_amdgcn_s_wait_asynccnt(0);   hist16(48);
#else
  for (int n = tid; n < NN; n += THREADS)
    data[n] = __float_as_uint(x[base + (size_t)n * DD]);
  __syncthreads();
  hist16(0); hist16(16); hist16(32); hist16(48);
#endif
  __syncthreads();

  // ---------------- Phase C: locate level-1 bucket per target rank ----------
  if (tid == 0) {
    int Kr[NRANK];
    #pragma clang loop unroll(disable)
    for (int r = 0; r < NPCT; ++r) {
      float fidx = (0.05f + 0.1f * (float)r) * (float)(NN - 1);
      int lo = (int)floorf(fidx);
      Kr[2 * r]     = lo;
      Kr[2 * r + 1] = lo + 1;   // fidx is never integral for these fracs
    }
    int ns = 0;
    #pragma clang loop unroll(disable)
    for (int k = 0; k < NRANK; ++k) {
      int kk = Kr[k], acc = 0, b1 = 255, loc = 0;
      #pragma clang loop unroll(disable)
      for (int bb = 0; bb < 256; ++bb) {
        int h = (int)H1[bb];
        if (kk < acc + h) { b1 = bb; loc = kk - acc; break; }
        acc += h;
      }
      if (ns == 0 || bucket1OfSlot[ns - 1] != b1) bucket1OfSlot[ns++] = b1;
      slotOfRank[k] = ns - 1;
      rank1[k]      = loc;
    }
    #pragma clang loop unroll(disable)
    for (int i = 0; i < 256; ++i) map1[i] = -1;
    #pragma clang loop unroll(disable)
    for (int s = 0; s < ns; ++s)  map1[bucket1OfSlot[s]] = s;
  }
  __syncthreads();

  // ---------------- Phase D: level-2 histograms (key[23:16]) ----------------
  for (int i = tid; i < NRANK * 256; i += THREADS) H2[i] = 0u;
  __syncthreads();
  #pragma clang loop unroll_count(4)
  for (int n = tid; n < NN; n += THREADS) {
    unsigned key = keyOf(data[n]);
    int s = map1[key >> 24];
    if (s >= 0) atomicAdd(&H2[s * 256 + ((key >> 16) & 255u)], 1u);
  }
  __syncthreads();

  // ---------------- Phase E: 16-bit prefix + local rank per target ----------
  if (tid == 0) {
    int ns2 = 0;
    #pragma clang loop unroll(disable)
    for (int k = 0; k < NRANK; ++k) {
      int s = slotOfRank[k];
      int kk = rank1[k], acc = 0, b2 = 255, loc = 0;
      #pragma clang loop unroll(disable)
      for (int bb = 0; bb < 256; ++bb) {
        int h = (int)H2[s * 256 + bb];
        if (kk < acc + h) { b2 = bb; loc = kk - acc; break; }
        acc += h;
      }
      unsigned p = (((unsigned)bucket1OfSlot[s]) << 8) | (unsigned)b2;
      if (ns2 == 0 || prefix2[ns2 - 1] != p) prefix2[ns2++] = p;
      slot2OfRank[k] = ns2 - 1;
      rank2[k]       = loc;
    }
    nslot2 = ns2;
  }
  __syncthreads();

  // ---------------- Phase F: collect candidates per 16-bit prefix -----------
  const int ns2 = nslot2;
  #pragma clang loop unroll_count(2)
  for (int n = tid; n < NN; n += THREADS) {
    unsigned key = keyOf(data[n]);
    unsigned p = key >> 16;
    #pragma clang loop unroll(disable)
    for (int j = 0; j < ns2; ++j) {
      if (p == prefix2[j]) {
        unsigned pos = atomicAdd(&cnt2[j], 1u);
        if (pos < CAP) cand[j * CAP + pos] = key;
        break;
      }
    }
  }
  __syncthreads();

  // ---------------- Phase G: exact selection inside tiny candidate sets -----
  #pragma clang loop unroll(disable)
  for (int j = 0; j < ns2; ++j) {
    int cnt = (int)cnt2[j]; if (cnt > CAP) cnt = CAP;
    #pragma clang loop unroll(disable)
    for (int i = tid; i < cnt; i += THREADS) {
      unsigned c = cand[j * CAP + i];
      int rk = 0;
      #pragma clang loop unroll_count(4)
      for (int m = 0; m < cnt; ++m) {
        unsigned cm = cand[j * CAP + m];
        rk += (cm < c) | ((cm == c) & (m < i));   // stable rank
      }
      #pragma clang loop unroll(disable)
      for (int k = 0; k < NRANK; ++k)
        if (slot2OfRank[k] == j && rank2[k] == rk) keyOut[k] = c;
    }
  }
  __syncthreads();

  // ---------------- Phase H: lerp + write (out[b][d*10 + r]) ----------------
  if (tid < NPCT) {
    int r = tid;
    float fidx = (0.05f + 0.1f * (float)r) * (float)(NN - 1);
    int   lo   = (int)floorf(fidx);
    float w    = fidx - (float)lo;
    float vlo  = fromKey(keyOut[2 * r]);
    float vhi  = fromKey(keyOut[2 * r + 1]);
    out[(size_t)b * (DD * NPCT) + (size_t)d * NPCT + (size_t)r] =
        vlo * (1.0f - w) + vhi * w;
  }
}

extern "C" void kernel_launch(void* const* d_in, const int* in_sizes, int n_in,
                              void* d_out, int out_size, void* d_ws, size_t ws_size,
                              hipStream_t stream) {
  (void)in_sizes; (void)n_in; (void)d_ws; (void)ws_size; (void)out_size;
  const float* x = (const float*)d_in[0];
  float* out = (float*)d_out;
  dim3 grid(BB * DD);      // 4096 columns, one workgroup each
  dim3 block(THREADS);
  hipLaunchKernelGGL(PercentileAggregator_kernel, grid, block, 0, stream, x, out);
}